// ExpertCombiner_54546084659701
// MI455X (gfx1250) — compile-verified
//
#include <hip/hip_runtime.h>

// ExpertCombiner for MI455X (gfx1250).
//
// out[tok, :] += w[e,c] * x[e,c, :]  with tok = idx[e,c]; random indices =>
// collisions guaranteed => scatter must use atomics.
//
// Roofline: 256 MiB one-shot read (expert rows) + 128 MiB output. Output fits
// in the 192 MB global L2, so: zero it with RT B128 stores (lines allocate and
// stay in L2), stream expert rows with NT B128 loads (do not evict the output
// working set), and let the 67M non-returning global_atomic_add_f32 RMWs
// resolve at L2. HBM floor ~20 us @ 23.3 TB/s. No matmul here -> WMMA N/A;
// the CDNA5 levers are the TH cache hints, wave32 block sizing, and scalar
// (s_load) handling of block-uniform metadata.

typedef __attribute__((ext_vector_type(4))) float v4f;

constexpr int kD       = 1024;        // feature dim per row
constexpr int kThreads = kD / 4;      // 256 threads = 8 wave32s; one float4 per lane

// ---------------------------------------------------------------------------
// Scatter: one block per (e,c) row. Row weight + token index are block-
// uniform -> scalar loads. Each lane: NT 128-bit load of 4 elements, scale,
// 4 non-returning global_atomic_add_f32 into the L2-resident output.
// (Defined first so the disasm snippet shows this kernel.)
// ---------------------------------------------------------------------------
__global__ __launch_bounds__(kThreads) void expert_scatter_add(
    const float* __restrict__ x,     // [E*C, D] expert_outputs
    const float* __restrict__ w,     // [E*C]    weights
    const int*   __restrict__ idx,   // [E*C]    token indices in [0, B*S)
    float*       __restrict__ out)   // [B*S, D]
{
    const unsigned row    = blockIdx.x;          // uniform
    const float    weight = w[row];              // uniform -> s_load_b32
    const unsigned tok    = (unsigned)idx[row];  // uniform -> s_load_b32

    const v4f* src = (const v4f*)(x + (size_t)row * kD);
    // Non-temporal: this 256 MiB stream is touched exactly once; keep it out
    // of L2 so the 128 MiB output working set stays resident.
    v4f v = __builtin_nontemporal_load(src + threadIdx.x);

    // Scale once; compiler can dual-issue as v_dual_mul_f32 pairs.
    v.x *= weight;
    v.y *= weight;
    v.z *= weight;
    v.w *= weight;

    float* dst = out + (size_t)tok * kD + (size_t)threadIdx.x * 4;
    // Return value unused -> non-returning global_atomic_add_f32 (STOREcnt),
    // default RT temporal hint keeps the line hot in L2.
    atomicAdd(dst + 0, v.x);
    atomicAdd(dst + 1, v.y);
    atomicAdd(dst + 2, v.z);
    atomicAdd(dst + 3, v.w);
}

// ---------------------------------------------------------------------------
// Zero pass: 128-bit RT stores allocate the output in L2 for the atomic phase.
// ---------------------------------------------------------------------------
__global__ __launch_bounds__(kThreads) void zero_out_f4(v4f* __restrict__ out, int n4) {
    int i = blockIdx.x * kThreads + threadIdx.x;
    if (i < n4) {
        v4f z = {0.0f, 0.0f, 0.0f, 0.0f};
        out[i] = z;                       // global_store_b128 (RT)
    }
}

// ---------------------------------------------------------------------------
// Launch: zero pass then scatter pass, both on `stream` (in-order).
// ---------------------------------------------------------------------------
extern "C" void kernel_launch(void* const* d_in, const int* in_sizes, int n_in,
                              void* d_out, int out_size, void* d_ws, size_t ws_size,
                              hipStream_t stream) {
    const float* x   = (const float*)d_in[0];  // expert_outputs [E, C, D] f32
    const float* w   = (const float*)d_in[1];  // weights        [E, C]    f32
    const int*   idx = (const int*)  d_in[2];  // token_indices  [E, C]    int
    float*       out = (float*)d_out;          // [B, S, D] f32

    const int n_rows = in_sizes[1];            // E*C (= weights element count)
    const int n4     = out_size / 4;           // output float4 count

    const int zero_blocks = (n4 + kThreads - 1) / kThreads;
    zero_out_f4<<<zero_blocks, kThreads, 0, stream>>>((v4f*)out, n4);

    expert_scatter_add<<<n_rows, kThreads, 0, stream>>>(x, w, idx, out);
}